// TensorBoard_85890755985905
// MI455X (gfx1250) — compile-verified
//
#include <hip/hip_runtime.h>
#include <hip/hip_bf16.h>
#include <stdint.h>

// Problem constants (match the reference harness exactly)
#define NB    1024   // boards
#define NN2   361    // board cells / candidate moves
#define MH    361    // history length
#define MHPAD 364    // MH rounded up to a multiple of 4 (for int4 scans)
#define MAXG  64     // max groups per board staged in LDS (reference uses 8)

#define AS_GLOBAL __attribute__((address_space(1)))
#define AS_LDS    __attribute__((address_space(3)))

#if defined(__gfx1250__) && __has_builtin(__builtin_amdgcn_global_load_async_to_lds_b32)
#define HAVE_ASYNC_LDS 1
#else
#define HAVE_ASYNC_LDS 0
#endif

__global__ __launch_bounds__(384)
void superko_mask_kernel(const int* __restrict__ ZposT,        // [3, N2]
                         const int* __restrict__ cur_player,   // [B]
                         const int* __restrict__ cur_hash,     // [B]
                         const int* __restrict__ hist,         // [B, M]
                         const int* __restrict__ mcount,       // [B]
                         const unsigned char* __restrict__ legal, // [B, N2] bool
                         const int* __restrict__ sgi,          // [K] stone -> cell
                         const int* __restrict__ sptr,         // [R+1] group -> stone range
                         const int* __restrict__ gpb,          // [B+1] board -> group range
                         const int* __restrict__ cap,          // [B, N2, 4] local group ids or -1
                         float* __restrict__ out)               // [B, N2]
{
    __shared__ int s_hist[MHPAD];  // this board's hash history (async DMA target; tail masked)
    __shared__ int s_gx[MAXG];     // per-local-group removal XOR delta

    const int b = blockIdx.x;
    const int t = threadIdx.x;

    // Uniform per-board scalars
    const int player  = cur_player[b];
    const int opp_row = (2 - player) * NN2;   // ZposT row offset of opponent color (1 + (1-player))
    const int chash   = cur_hash[b];
    const int mc      = mcount[b];

    // ---- Phase 1a: async-stage this board's history row into LDS (CDNA5 ASYNCcnt path) ----
    // Rows are only 4B-aligned (1444B pitch), so stage at B32 granularity.
    const int* hrow = hist + (size_t)b * MH;
    if (t < MH) {
#if HAVE_ASYNC_LDS
        __builtin_amdgcn_global_load_async_to_lds_b32(
            (AS_GLOBAL int*)(hrow + t),
            (AS_LDS int*)(&s_hist[t]),
            /*offset=*/0, /*cpol=*/0);
#else
        s_hist[t] = hrow[t];
#endif
    }

    // ---- Phase 1b (overlapped with DMA): segmented XOR of per-stone Zobrist deltas ----
    // group_xor[g] = XOR_{stones s in g} ( Z[1+opp][cell(s)] ^ Z[0][cell(s)] )
    const int gbeg = gpb[b];
    const int gnum = gpb[b + 1] - gbeg;
    if (t < gnum && t < MAXG) {
        const int g  = gbeg + t;
        const int s0 = sptr[g];
        const int s1 = sptr[g + 1];
        int acc = 0;
        for (int s = s0; s < s1; ++s) {
            const int idx = sgi[s];
            acc ^= ZposT[opp_row + idx] ^ ZposT[idx];   // z_opp ^ z_empty
        }
        s_gx[t] = acc;
    }

    // ---- Fence DMA + make LDS visible block-wide ----
#if HAVE_ASYNC_LDS
#if __has_builtin(__builtin_amdgcn_s_wait_asynccnt)
    __builtin_amdgcn_s_wait_asynccnt(0);
#else
    asm volatile("s_wait_asynccnt 0" ::: "memory");
#endif
#endif
    __syncthreads();

    // ---- Phase 2: one lane per candidate cell ----
    if (t < NN2) {
        const int z_empty = ZposT[t];
        const int z_color = ZposT[(1 + player) * NN2 + t];
        int cand = chash ^ z_empty ^ z_color;           // placement delta folded in

        // 4 capture slots: 16B-aligned row -> single global_load_b128
        const int4 c4 = *(const int4*)(cap + ((size_t)b * NN2 + t) * 4);
        if (c4.x >= 0) cand ^= s_gx[c4.x];
        if (c4.y >= 0) cand ^= s_gx[c4.y];
        if (c4.z >= 0) cand ^= s_gx[c4.z];
        if (c4.w >= 0) cand ^= s_gx[c4.w];

        // History scan from LDS (uniform address -> broadcast reads).
        // Main loop: full quads, no masking. Tail: one masked quad (mc is wave-uniform,
        // so the tail test is a uniform scalar branch; the mask also excludes the
        // uninitialized LDS padding words).
        int rep = 0;
        const int nfull = mc >> 2;                       // scalar-uniform trip count
        const int4* h4 = (const int4*)s_hist;
        for (int j4 = 0; j4 < nfull; ++j4) {
            const int4 h = h4[j4];                       // ds_load_b128, 16B aligned
            rep |= (h.x == cand) | (h.y == cand) | (h.z == cand) | (h.w == cand);
        }
        const int trem = mc - (nfull << 2);              // 0..3, uniform
        if (trem > 0) {
            const int4 h = h4[nfull];
            rep |= (h.x == cand);
            rep |= (trem > 1) & (h.y == cand);
            rep |= (trem > 2) & (h.z == cand);
        }

        const bool ok = (legal[(size_t)b * NN2 + t] != 0) && (rep == 0);
        out[(size_t)b * NN2 + t] = ok ? 1.0f : 0.0f;
    }
}

extern "C" void kernel_launch(void* const* d_in, const int* in_sizes, int n_in,
                              void* d_out, int out_size, void* d_ws, size_t ws_size,
                              hipStream_t stream) {
    (void)n_in; (void)d_ws; (void)ws_size; (void)out_size;

    // setup_inputs() order:
    const int* ZposT        = (const int*)d_in[0];           // [3, N2]
    const int* cur_player   = (const int*)d_in[1];           // [B]
    const int* cur_hash     = (const int*)d_in[2];           // [B]
    const int* hist         = (const int*)d_in[3];           // [B, M]
    const int* mcount       = (const int*)d_in[4];           // [B]
    const unsigned char* lm = (const unsigned char*)d_in[5]; // [B, N2] bool
    const int* sgi          = (const int*)d_in[6];           // [K]
    const int* sptr         = (const int*)d_in[7];           // [R+1]
    const int* gpb          = (const int*)d_in[8];           // [B+1]
    const int* cap          = (const int*)d_in[9];           // [B, N2, 4]
    // d_in[10] = scale (unused by reference output)

    float* out = (float*)d_out;

    const int b = in_sizes[1];  // number of boards
    superko_mask_kernel<<<b, 384, 0, stream>>>(ZposT, cur_player, cur_hash, hist, mcount,
                                               lm, sgi, sptr, gpb, cap, out);
}